// LengthRegulator_59399397704210
// MI455X (gfx1250) — compile-verified
//
#include <hip/hip_runtime.h>
#include <stdint.h>

#define MAX_DUR 10000
#define B_   32
#define C_   384
#define T_   1024
#define ML   8192
#define OUTC 386   // C + idx_in + length
#define CPB  8     // channels per expand block (8 waves x 1 channel)

typedef unsigned int u32x4 __attribute__((ext_vector_type(4)));
typedef int          i32x8 __attribute__((ext_vector_type(8)));
typedef int          i32x4 __attribute__((ext_vector_type(4)));

#if __has_builtin(__builtin_amdgcn_tensor_load_to_lds) && __has_builtin(__builtin_amdgcn_s_wait_tensorcnt)
#define HAVE_TDM 1
#else
#define HAVE_TDM 0
#endif

// Build a 2D tensor-DMA descriptor (D# groups 0/1 per CDNA5 ISA §8) and issue
// TENSOR_LOAD_TO_LDS. data_size = 4 bytes. Must be executed wave-uniformly.
// clang-23 lane: 6-arg builtin (g0, g1, g2, g3, g-ext, cpol).
__device__ __forceinline__ void tdm_load_2d(const void* gptr, uint32_t lds_off,
                                            uint32_t tensor_d0, uint32_t tensor_d1,
                                            uint32_t tile_d0, uint32_t tile_d1,
                                            uint32_t stride0) {
#if HAVE_TDM
  uint64_t ga = (uint64_t)(uintptr_t)gptr;
  u32x4 g0;
  g0.x = 1u;                                                   // count=1 (valid user D#)
  g0.y = lds_off;                                              // lds_addr (bytes)
  g0.z = (uint32_t)(ga & 0xFFFFFFFFull);                       // global_addr[31:0]
  g0.w = (uint32_t)((ga >> 32) & 0x1FFFFFFull) | (2u << 30);   // global_addr[56:32], type=2
  i32x8 g1;
  g1[0] = (int)(2u << 16);                                     // data_size=2 (4 bytes)
  g1[1] = (int)(tensor_d0 << 16);                              // tensor_dim0[15:0]
  g1[2] = (int)((tensor_d0 >> 16) | (tensor_d1 << 16));        // tensor_dim0[31:16] | tensor_dim1[15:0]
  g1[3] = (int)((tensor_d1 >> 16) | (tile_d0 << 16));          // tensor_dim1[31:16] | tile_dim0
  g1[4] = (int)(tile_d1 & 0xFFFFu);                            // tile_dim1 (tile_dim2=0)
  g1[5] = (int)stride0;                                        // tensor_dim0_stride[31:0]
  g1[6] = 0;                                                   // stride0[47:32]=0, dim1_stride lo=0
  g1[7] = 0;
  i32x4 z4 = {0, 0, 0, 0};
  i32x8 z8 = {0, 0, 0, 0, 0, 0, 0, 0};
  __builtin_amdgcn_tensor_load_to_lds(g0, g1, z4, z4, z8, 0);
#else
  (void)gptr; (void)lds_off; (void)tensor_d0; (void)tensor_d1;
  (void)tile_d0; (void)tile_d1; (void)stride0;
#endif
}

// ---------------------------------------------------------------------------
// Kernel 1: per-batch duration scan + token-map scatter + rows 384/385 + mel_len
// ---------------------------------------------------------------------------
__global__ __launch_bounds__(1024) void lr_scan_kernel(
    const long long* __restrict__ dur, int* __restrict__ tokmap,
    float* __restrict__ out, float* __restrict__ mel) {
  __shared__ int s[T_];
  const int b = blockIdx.x;
  const int t = threadIdx.x;

  long long dl = dur[(size_t)b * T_ + t];
  if (dl < 0) dl = -dl;
  int d = (int)dl;
  if (d < 1) d = 1;
  if (d > MAX_DUR) d = MAX_DUR;

  s[t] = d;
  __syncthreads();
  // Hillis-Steele inclusive scan over 1024 tokens
  #pragma unroll
  for (int off = 1; off < T_; off <<= 1) {
    int v = (t >= off) ? s[t - off] : 0;
    __syncthreads();
    s[t] += v;
    __syncthreads();
  }
  const int end   = s[t];
  const int start = end - d;
  const int total = s[T_ - 1];

  int*   tm   = tokmap + (size_t)b * ML;
  float* oidx = out + ((size_t)b * OUTC + C_) * ML;      // row 384: idx_in
  float* olen = out + ((size_t)b * OUTC + C_ + 1) * ML;  // row 385: length

  for (int j = 0; j < d; ++j) {
    int p = start + j;
    if (p < ML) {
      tm[p]   = t;
      oidx[p] = (float)j;
      olen[p] = (float)d;
    }
  }
  // Invalid tail: sentinel token + zeroed aux rows
  for (int p = total + t; p < ML; p += T_) {
    tm[p]   = -1;
    oidx[p] = 0.0f;
    olen[p] = 0.0f;
  }
  if (t == 0) mel[b] = (float)total;
}

// ---------------------------------------------------------------------------
// Kernel 2: expand 8 channels per block (8 waves). TDM stages tokmap (32KB) +
// 8 x-rows (32KB) into LDS; each wave gathers one channel and stores float4.
// ---------------------------------------------------------------------------
__global__ __launch_bounds__(256) void lr_expand_kernel(
    const float* __restrict__ x, const int* __restrict__ tokmap,
    float* __restrict__ out) {
  __shared__ int   tok_s[ML];          // 32 KB
  __shared__ float xrow_s[CPB * T_];   // 32 KB
  const int b  = blockIdx.y;
  const int c0 = blockIdx.x * CPB;

#if HAVE_TDM
  if (threadIdx.x < 32) {  // wave 0 issues both DMA descriptors
    tdm_load_2d(tokmap + (size_t)b * ML,
                (uint32_t)(uintptr_t)(void*)tok_s,
                ML, 1, ML, 1, ML);
    tdm_load_2d(x + ((size_t)b * C_ + c0) * T_,
                (uint32_t)(uintptr_t)(void*)xrow_s,
                T_, CPB, T_, CPB, T_);
    __builtin_amdgcn_s_wait_tensorcnt(0);
  }
#else
  {
    const int4* tsrc = (const int4*)(tokmap + (size_t)b * ML);
    int4*       tdst = (int4*)tok_s;
    for (int i = threadIdx.x; i < ML / 4; i += 256) tdst[i] = tsrc[i];
    const float4* xsrc = (const float4*)(x + ((size_t)b * C_ + c0) * T_);
    float4*       xdst = (float4*)xrow_s;
    for (int i = threadIdx.x; i < (CPB * T_) / 4; i += 256) xdst[i] = xsrc[i];
  }
#endif
  __syncthreads();

  const int wave = threadIdx.x >> 5;
  const int lane = threadIdx.x & 31;
  const int c    = c0 + wave;
  float* __restrict__ orow = out + ((size_t)b * OUTC + c) * ML;
  const float* xr = xrow_s + wave * T_;

  #pragma unroll 4
  for (int p = lane * 4; p < ML; p += 128) {
    int4 tk = *(const int4*)(tok_s + p);
    float4 v;
    v.x = (tk.x >= 0) ? xr[tk.x] : 0.0f;
    v.y = (tk.y >= 0) ? xr[tk.y] : 0.0f;
    v.z = (tk.z >= 0) ? xr[tk.z] : 0.0f;
    v.w = (tk.w >= 0) ? xr[tk.w] : 0.0f;
    *(float4*)(orow + p) = v;   // coalesced b128 store
  }
}

extern "C" void kernel_launch(void* const* d_in, const int* in_sizes, int n_in,
                              void* d_out, int out_size, void* d_ws, size_t ws_size,
                              hipStream_t stream) {
  const float*     x   = (const float*)d_in[0];
  const long long* dur = (const long long*)d_in[1];  // int64 durations
  // d_in[2] = max_len (8192), fixed at compile time
  float* out    = (float*)d_out;
  float* mel    = out + (size_t)B_ * OUTC * ML;      // mel_len after out, flat
  int*   tokmap = (int*)d_ws;                        // 32 * 8192 * 4 = 1 MB

  lr_scan_kernel<<<dim3(B_), dim3(1024), 0, stream>>>(dur, tokmap, out, mel);
  lr_expand_kernel<<<dim3(C_ / CPB, B_), dim3(256), 0, stream>>>(x, tokmap, out);
}